// SimpleGNN_91018946937353
// MI455X (gfx1250) — compile-verified
//
#include <hip/hip_runtime.h>

typedef __attribute__((ext_vector_type(2))) float v2f;
typedef __attribute__((ext_vector_type(8))) float v8f;

// ---------------------------------------------------------------------------
// Elementwise helpers
// ---------------------------------------------------------------------------
__global__ void fill_f32(float* __restrict__ p, float v, long long n) {
    long long i = (long long)blockIdx.x * blockDim.x + threadIdx.x;
    if (i < n) p[i] = v;
}

__global__ void deg_count(const int* __restrict__ dst, float* __restrict__ deg, int ne) {
    int e = blockIdx.x * blockDim.x + threadIdx.x;
    if (e < ne) atomicAdd(&deg[dst[e]], 1.0f);
}

__global__ void rsqrt_inplace(float* __restrict__ d, int n) {
    int i = blockIdx.x * blockDim.x + threadIdx.x;
    if (i < n) {
        float v = d[i];
        d[i] = rsqrtf(v > 1.0f ? v : 1.0f);
    }
}

// ---------------------------------------------------------------------------
// Exact-fp32 WMMA GEMM:  C[nrows x 64] = A[nrows x 64] * W[64 x 64]
// One wave -> one 16x16 output tile, accumulated over K=64 via 16 WMMAs of K=4.
// Block = 256 threads = 8 waves = 2 row-tiles x 4 col-tiles.
// ---------------------------------------------------------------------------
__global__ void __launch_bounds__(256) gemm64_wmma_f32(
    const float* __restrict__ A, const float* __restrict__ W,
    float* __restrict__ C, int nrows)
{
    __shared__ float Wlds[64 * 64];
    for (int i = threadIdx.x; i < 64 * 64; i += 256) Wlds[i] = W[i];
    __syncthreads();

    const int wave = threadIdx.x >> 5;       // 0..7
    const int lane = threadIdx.x & 31;
    const int rt   = wave >> 2;              // 0..1  (row tile in block)
    const int ct   = wave & 3;               // 0..3  (col tile: 4 x 16 = 64)
    const int row_tile = (blockIdx.x * 2 + rt) * 16;
    if (row_tile + 16 > nrows) return;       // nrows is a multiple of 16 here
    const int col0  = ct * 16;
    const int lhalf = lane >> 4;             // 0/1 -> K pair select
    const int l15   = lane & 15;

    const float* arow = A + (long long)(row_tile + l15) * 64;

    v8f acc = {};
#pragma unroll
    for (int k0 = 0; k0 < 64; k0 += 4) {
        const int kk = k0 + 2 * lhalf;
        v2f a, b;
        a.x = arow[kk];
        a.y = arow[kk + 1];
        b.x = Wlds[kk * 64 + col0 + l15];
        b.y = Wlds[(kk + 1) * 64 + col0 + l15];
        acc = __builtin_amdgcn_wmma_f32_16x16x4_f32(
            /*neg_a=*/false, a, /*neg_b=*/false, b,
            /*c_mod=*/(short)0, acc, /*reuse_a=*/false, /*reuse_b=*/false);
    }
#pragma unroll
    for (int r = 0; r < 8; ++r) {
        C[(long long)(row_tile + r + 8 * lhalf) * 64 + col0 + l15] = acc[r];
    }
}

// ---------------------------------------------------------------------------
// Edge-parallel scatter: agg[dst] += h[src] * dinv[src]*dinv[dst]
// Each thread handles one edge x one float2 feature pair (32 threads/edge).
// All traffic is L2-resident (h/agg = 25.6 MB each, L2 = 192 MB).
// ---------------------------------------------------------------------------
__global__ void mp_scatter(const float* __restrict__ h, const float* __restrict__ dinv,
                           const int* __restrict__ src, const int* __restrict__ dst,
                           float* __restrict__ agg, int ne)
{
    long long idx = (long long)blockIdx.x * blockDim.x + threadIdx.x;
    long long total = (long long)ne * 32;
    if (idx >= total) return;
    int e = (int)(idx >> 5);
    int f = (int)(idx & 31) * 2;
    int s = src[e], d = dst[e];
    float nrm = dinv[s] * dinv[d];
    float2 hv = *(const float2*)(h + (long long)s * 64 + f);
    atomicAdd(&agg[(long long)d * 64 + f],     hv.x * nrm);
    atomicAdd(&agg[(long long)d * 64 + f + 1], hv.y * nrm);
}

// out[i,f] = relu(agg[i,f] + h[i,f]*dinv[i]^2 + b[f])   (self-loop folded in)
__global__ void bias_relu_self(const float* agg, const float* __restrict__ h,
                               const float* __restrict__ dinv, const float* __restrict__ b,
                               float* out, int n)
{
    long long idx = (long long)blockIdx.x * blockDim.x + threadIdx.x;
    if (idx >= (long long)n * 64) return;
    int i = (int)(idx >> 6);
    int f = (int)(idx & 63);
    float di = dinv[i];
    float v = agg[idx] + h[idx] * di * di + b[f];
    out[idx] = v > 0.0f ? v : 0.0f;
}

// ---------------------------------------------------------------------------
// Pooling + head
// ---------------------------------------------------------------------------
__global__ void pool_cnt(const int* __restrict__ batch, float* __restrict__ cnt, int n) {
    int i = blockIdx.x * blockDim.x + threadIdx.x;
    if (i < n) atomicAdd(&cnt[batch[i]], 1.0f);
}

__global__ void pool_sum(const float* __restrict__ act, const int* __restrict__ batch,
                         float* __restrict__ pooled, int n)
{
    long long idx = (long long)blockIdx.x * blockDim.x + threadIdx.x;
    if (idx >= (long long)n * 64) return;
    int i = (int)(idx >> 6);
    int f = (int)(idx & 63);
    atomicAdd(&pooled[(long long)batch[i] * 64 + f], act[idx]);
}

__global__ void head_linear(const float* __restrict__ pooled, const float* __restrict__ cnt,
                            const float* __restrict__ Wl, const float* __restrict__ bl,
                            float* __restrict__ out, int g_times_o, int nout)
{
    int idx = blockIdx.x * blockDim.x + threadIdx.x;
    if (idx >= g_times_o) return;
    int g = idx / nout, o = idx % nout;
    float c = cnt[g]; c = c > 1.0f ? c : 1.0f;
    float s = 0.0f;
#pragma unroll
    for (int k = 0; k < 64; ++k) s += pooled[g * 64 + k] * Wl[k * nout + o];
    out[idx] = s / c + bl[o];
}

// ---------------------------------------------------------------------------
// Launch
// ---------------------------------------------------------------------------
extern "C" void kernel_launch(void* const* d_in, const int* in_sizes, int n_in,
                              void* d_out, int out_size, void* d_ws, size_t ws_size,
                              hipStream_t stream)
{
    const float* x     = (const float*)d_in[0];
    const int*   eidx  = (const int*)d_in[1];
    const int*   batch = (const int*)d_in[2];
    const float* W1    = (const float*)d_in[3];
    const float* b1    = (const float*)d_in[4];
    const float* W2    = (const float*)d_in[5];
    const float* b2    = (const float*)d_in[6];
    const float* Wl    = (const float*)d_in[7];
    const float* bl    = (const float*)d_in[8];
    float* out = (float*)d_out;

    const int N = in_sizes[2];           // nodes
    const int E = in_sizes[1] / 2;       // edges
    const int NOUT = in_sizes[8];        // 10
    const int G = out_size / NOUT;       // 256
    const int* src = eidx;
    const int* dst = eidx + E;

    // Workspace layout (floats)
    float* dinv   = (float*)d_ws;                     // N
    float* hbuf   = dinv + ((N + 127) & ~127);        // N*64
    float* aggbuf = hbuf + (long long)N * 64;         // N*64
    float* pooled = aggbuf + (long long)N * 64;       // G*64
    float* cnt    = pooled + (long long)G * 64;       // G

    const int T = 256;
    const long long NF = (long long)N * 64;
    dim3 gN((N + T - 1) / T), gE((E + T - 1) / T);
    dim3 gNF((unsigned)((NF + T - 1) / T));
    dim3 gEF((unsigned)(((long long)E * 32 + T - 1) / T));
    dim3 gGemm((N + 31) / 32);

    // degrees: 1 (self loop) + in-degree, then dinv = rsqrt
    fill_f32<<<gN, T, 0, stream>>>(dinv, 1.0f, N);
    deg_count<<<gE, T, 0, stream>>>(dst, dinv, E);
    rsqrt_inplace<<<gN, T, 0, stream>>>(dinv, N);

    // ---- Layer 1 ----
    gemm64_wmma_f32<<<gGemm, 256, 0, stream>>>(x, W1, hbuf, N);
    fill_f32<<<gNF, T, 0, stream>>>(aggbuf, 0.0f, NF);
    mp_scatter<<<gEF, T, 0, stream>>>(hbuf, dinv, src, dst, aggbuf, E);
    bias_relu_self<<<gNF, T, 0, stream>>>(aggbuf, hbuf, dinv, b1, aggbuf, N);

    // ---- Layer 2 ----  (act1 lives in aggbuf; overwrite hbuf with h2)
    gemm64_wmma_f32<<<gGemm, 256, 0, stream>>>(aggbuf, W2, hbuf, N);
    fill_f32<<<gNF, T, 0, stream>>>(aggbuf, 0.0f, NF);
    mp_scatter<<<gEF, T, 0, stream>>>(hbuf, dinv, src, dst, aggbuf, E);
    bias_relu_self<<<gNF, T, 0, stream>>>(aggbuf, hbuf, dinv, b2, aggbuf, N);

    // ---- Pool + head ----
    fill_f32<<<dim3((G * 64 + T - 1) / T), T, 0, stream>>>(pooled, 0.0f, (long long)G * 64);
    fill_f32<<<dim3((G + T - 1) / T), T, 0, stream>>>(cnt, 0.0f, G);
    pool_cnt<<<gN, T, 0, stream>>>(batch, cnt, N);
    pool_sum<<<gNF, T, 0, stream>>>(aggbuf, batch, pooled, N);
    head_linear<<<dim3((G * NOUT + T - 1) / T), T, 0, stream>>>(pooled, cnt, Wl, bl, out, G * NOUT, NOUT);
}